// ArbitraryODE_66795331387937
// MI455X (gfx1250) — compile-verified
//
#include <hip/hip_runtime.h>
#include <hip/hip_bf16.h>
#include <stdint.h>

// ---------------------------------------------------------------------------
// Graph message passing: per-edge coefficient (exp/tanh mix) * dpos, scatter
// summed by dst, normalized by valid-edge counts.
//   d_in[0] pos        (N,2)  f32
//   d_in[1] p          (4,4)  f32
//   d_in[2] cell_type  (N,)   i32
//   d_in[3] edge_index (2,E)  i32   row0 = dst, row1 = src
//   d_in[4] func_type  (4,)   i32
//   d_out              (N,2)  f32
// CDNA5 path: double-buffered global_load_async_to_lds_b128 streaming of the
// edge-index array (ASYNCcnt / s_wait_asynccnt), native trans ops for
// exp2/log2/tanh/rsq, global_atomic_add_f32 scatter.
// ---------------------------------------------------------------------------

#define TPB       256
#define TILE      1024          // edges per tile; 2 async b128 instrs per wave
#define GRID_EDGE 512

#define SIGMA     0.05f
// inv_2s2 * log2(e): exp(-x*inv_2s2) == exp2(-x * C_EXP)
#define C_EXP     (1.4426950408889634f / (2.0f * SIGMA * SIGMA))

#if __has_builtin(__builtin_amdgcn_tanhf)
#define TANHF __builtin_amdgcn_tanhf
#elif __has_builtin(__builtin_amdgcn_tanh_f32)
#define TANHF __builtin_amdgcn_tanh_f32
#else
#define TANHF tanhf
#endif

// v_exp_f32 computes 2^x; v_log_f32 computes log2(x)
#if __has_builtin(__builtin_amdgcn_exp2f)
#define EXP2F __builtin_amdgcn_exp2f
#else
#define EXP2F exp2f
#endif

#if __has_builtin(__builtin_amdgcn_logf)
#define LOG2F __builtin_amdgcn_logf
#else
#define LOG2F log2f
#endif

// Async copy of 16 bytes/lane from global (sgpr base + vgpr offset) into LDS.
__device__ __forceinline__ void async_copy_b128(uint32_t lds_off, uint64_t gbase,
                                                uint32_t goff) {
  asm volatile("global_load_async_to_lds_b128 %0, %1, %2 offset:0"
               :: "v"(lds_off), "v"(goff), "s"(gbase)
               : "memory");
}

__device__ __forceinline__ void process_edge(int dstv, int srcv,
                                             const float2* __restrict__ pos,
                                             const int* __restrict__ ctype,
                                             const float* sP, const int* sFT,
                                             float* __restrict__ sums,
                                             float* __restrict__ counts) {
  if (dstv == srcv) return;                    // invalid edge: contributes 0
  float2 pd = pos[dstv];
  float2 ps = pos[srcv];
  float dx = ps.x - pd.x;
  float dy = ps.y - pd.y;
  float d2 = fmaf(dx, dx, dy * dy);
  int ct = ctype[dstv];
  float4 pr = *(const float4*)(sP + 4 * ct);   // ds_load_b128
  int   isT = sFT[ct];

  // f1 = p0*exp(-d2^p1/2s^2) - p2*exp(-d2^p3/2s^2), via exp2/log2 trans unit
  float l2  = LOG2F(d2);
  float t1  = EXP2F(pr.y * l2);                // d2^p1
  float t3  = EXP2F(pr.w * l2);                // d2^p3
  float f1  = pr.x * EXP2F(-t1 * C_EXP) - pr.z * EXP2F(-t3 * C_EXP);

  // f2 = p0*tanh((dist-p1)*p2)/dist, dist = d2*rsq(d2)
  float rinv = __builtin_amdgcn_rsqf(d2);
  float dist = d2 * rinv;
  float f2   = pr.x * TANHF((dist - pr.y) * pr.z) * rinv;

  float coef = isT ? f2 : f1;
  atomicAdd(&sums[2 * dstv + 0], coef * dx);
  atomicAdd(&sums[2 * dstv + 1], coef * dy);
  atomicAdd(&counts[dstv], 1.0f);
}

__global__ void __launch_bounds__(TPB)
edge_kernel(const float2* __restrict__ pos, const float* __restrict__ p,
            const int* __restrict__ ctype, const int* __restrict__ eidx,
            const int* __restrict__ ftype, float* __restrict__ sums,
            float* __restrict__ counts, int E, int nTiles) {
  __shared__ __align__(16) int sIdx[2][2][TILE];  // [buf][dst|src][edge]
  __shared__ float sP[16];
  __shared__ int   sFT[4];

  const int tid = threadIdx.x;
  if (tid < 16) sP[tid] = p[tid];
  if (tid < 4)  sFT[tid] = ftype[tid] & 1;
  __syncthreads();

  const uint64_t dstBase = (uint64_t)(uintptr_t)eidx;            // row 0
  const uint64_t srcBase = (uint64_t)(uintptr_t)(eidx + E);      // row 1

  // Issue one tile's index stream (2 async instructions per wave).
  auto issue = [&](int tile, int buf) {
    uint32_t goff = (uint32_t)(tile * TILE + tid * 4) * 4u;      // byte offset
    uint32_t ldsD = (uint32_t)(uintptr_t)&sIdx[buf][0][tid * 4]; // LDS offset
    uint32_t ldsS = (uint32_t)(uintptr_t)&sIdx[buf][1][tid * 4];
    async_copy_b128(ldsD, dstBase, goff);
    async_copy_b128(ldsS, srcBase, goff);
  };

  int tile = blockIdx.x;
  int buf  = 0;
  if (tile < nTiles) issue(tile, 0);

  for (; tile < nTiles; tile += gridDim.x) {
    const int nxt = tile + gridDim.x;
    if (nxt < nTiles) {
      issue(nxt, buf ^ 1);
      // next tile's 2 asyncs may stay in flight; current tile's are done
      asm volatile("s_wait_asynccnt 0x2" ::: "memory");
    } else {
      asm volatile("s_wait_asynccnt 0x0" ::: "memory");
    }
    __syncthreads();

    const int4 d4 = *(const int4*)&sIdx[buf][0][tid * 4];
    const int4 s4 = *(const int4*)&sIdx[buf][1][tid * 4];
    process_edge(d4.x, s4.x, pos, ctype, sP, sFT, sums, counts);
    process_edge(d4.y, s4.y, pos, ctype, sP, sFT, sums, counts);
    process_edge(d4.z, s4.z, pos, ctype, sP, sFT, sums, counts);
    process_edge(d4.w, s4.w, pos, ctype, sP, sFT, sums, counts);

    __syncthreads();
    buf ^= 1;
  }

  // Ragged tail (empty for E = 3.2M since TILE | E, kept for safety).
  const int* dstRow = eidx;
  const int* srcRow = eidx + E;
  for (int e = nTiles * TILE + blockIdx.x * TPB + tid; e < E;
       e += gridDim.x * TPB) {
    process_edge(dstRow[e], srcRow[e], pos, ctype, sP, sFT, sums, counts);
  }
}

__global__ void zero_kernel(float* __restrict__ a, int na,
                            float* __restrict__ b, int nb) {
  const int stride = gridDim.x * blockDim.x;
  for (int i = blockIdx.x * blockDim.x + threadIdx.x; i < na; i += stride)
    a[i] = 0.0f;
  for (int i = blockIdx.x * blockDim.x + threadIdx.x; i < nb; i += stride)
    b[i] = 0.0f;
}

__global__ void finalize_kernel(float* __restrict__ out,
                                const float* __restrict__ counts, int N) {
  const int i = blockIdx.x * blockDim.x + threadIdx.x;
  if (i < N) {
    float c = fmaxf(counts[i], 1.0f);
    float r = 1.0f / c;
    float2 v = ((float2*)out)[i];
    v.x *= r;
    v.y *= r;
    ((float2*)out)[i] = v;
  }
}

extern "C" void kernel_launch(void* const* d_in, const int* in_sizes, int n_in,
                              void* d_out, int out_size, void* d_ws, size_t ws_size,
                              hipStream_t stream) {
  const float2* pos   = (const float2*)d_in[0];
  const float*  p     = (const float*)d_in[1];
  const int*    ctype = (const int*)d_in[2];
  const int*    eidx  = (const int*)d_in[3];
  const int*    ftype = (const int*)d_in[4];

  const int N = in_sizes[0] / 2;   // nodes
  const int E = in_sizes[3] / 2;   // edges

  float* sums   = (float*)d_out;   // accumulate directly into output buffer
  float* counts = (float*)d_ws;    // N floats of scratch

  zero_kernel<<<256, TPB, 0, stream>>>(sums, 2 * N, counts, N);

  const int nTiles = E / TILE;
  int grid = GRID_EDGE;
  if (nTiles > 0 && nTiles < grid) grid = nTiles;
  if (grid < 1) grid = 1;
  edge_kernel<<<grid, TPB, 0, stream>>>(pos, p, ctype, eidx, ftype,
                                        sums, counts, E, nTiles);

  finalize_kernel<<<(N + TPB - 1) / TPB, TPB, 0, stream>>>(sums, counts, N);
}